// SGCNet_10651518894449
// MI455X (gfx1250) — compile-verified
//
#include <hip/hip_runtime.h>
#include <math.h>

typedef float v2f __attribute__((ext_vector_type(2)));
typedef float v8f __attribute__((ext_vector_type(8)));

#define FDIM 300
#define FDIM4 75    // float4 chunks per row
#define NPAD 320    // padded column count (20 tiles of 16)

// ---------------- Phase 1: degrees & edge norms ----------------

__global__ void sgc_init_deg(int* __restrict__ deg, int n) {
    int i = blockIdx.x * blockDim.x + threadIdx.x;
    if (i < n) deg[i] = 1;   // self loop contributes 1
}

__global__ void sgc_count_deg(const int* __restrict__ col, int e, int* __restrict__ deg) {
    int i = blockIdx.x * blockDim.x + threadIdx.x;
    if (i < e) atomicAdd(&deg[col[i]], 1);
}

__global__ void sgc_dinv(const int* __restrict__ deg, float* __restrict__ dinv, int n) {
    int i = blockIdx.x * blockDim.x + threadIdx.x;
    if (i < n) dinv[i] = rsqrtf((float)deg[i]);   // deg >= 1 always
}

// Zero-padded copies of W (rows >= 300 are zero) and bias.
__global__ void sgc_pad(const float* __restrict__ W, const float* __restrict__ b,
                        float* __restrict__ Wp, float* __restrict__ bp) {
    int i = blockIdx.x * blockDim.x + threadIdx.x;
    if (i < NPAD * FDIM) {
        int nn = i / FDIM;                       // Wp and W share layout for nn < 300
        Wp[i] = (nn < FDIM) ? W[i] : 0.0f;
        if (i < NPAD) bp[i] = (i < FDIM) ? b[i] : 0.0f;
    }
}

// h[i,:] = x[i,:] * dinv[i]^2   (self-loop term; also initializes workspace)
__global__ void sgc_hinit(const float* __restrict__ x, const float* __restrict__ dinv,
                          float* __restrict__ h, int n) {
    int i = blockIdx.x * blockDim.x + threadIdx.x;   // over n * FDIM4 float4s
    if (i < n * FDIM4) {
        int r = i / FDIM4;
        float s = dinv[r];
        s = s * s;
        float4 v = ((const float4*)x)[i];
        v.x *= s; v.y *= s; v.z *= s; v.w *= s;
        ((float4*)h)[i] = v;
    }
}

// ---------------- Phase 2: edge scatter  h[c] += dinv[r]*dinv[c] * x[r] ----------------
// One wave32 per edge; float4 gathers; hardware f32 global atomics.

__global__ __launch_bounds__(256)
void sgc_scatter(const float* __restrict__ x, const int* __restrict__ row,
                 const int* __restrict__ col, const float* __restrict__ dinv,
                 float* __restrict__ h, int e) {
    int wid  = (blockIdx.x * blockDim.x + threadIdx.x) >> 5;
    int lane = threadIdx.x & 31;
    if (wid >= e) return;
    int r = row[wid];
    int c = col[wid];
    float w = dinv[r] * dinv[c];
    const float4* xr = (const float4*)x + (long)r * FDIM4;
    float* hc = h + (long)c * FDIM;
    for (int i = lane; i < FDIM4; i += 32) {
        float4 v = xr[i];
        float* p = hc + i * 4;
        __hip_atomic_fetch_add(p + 0, v.x * w, __ATOMIC_RELAXED, __HIP_MEMORY_SCOPE_AGENT);
        __hip_atomic_fetch_add(p + 1, v.y * w, __ATOMIC_RELAXED, __HIP_MEMORY_SCOPE_AGENT);
        __hip_atomic_fetch_add(p + 2, v.z * w, __ATOMIC_RELAXED, __HIP_MEMORY_SCOPE_AGENT);
        __hip_atomic_fetch_add(p + 3, v.w * w, __ATOMIC_RELAXED, __HIP_MEMORY_SCOPE_AGENT);
    }
}

// ---------------- Phase 3: out = l2norm(relu(h @ Wp^T + bp)) ----------------
// Block = 16-row M tile. 5 waves x 4 N-tiles of 16 cols = 320 padded cols.
// fp32 WMMA 16x16x4; branch-free inner loop (Wp zero-padded); fused LDS row norm.

__global__ __launch_bounds__(160)
void sgc_gemm_relu_norm(const float* __restrict__ h, const float* __restrict__ Wp,
                        const float* __restrict__ bp, float* __restrict__ out, int n) {
    __shared__ float rowsq[16];
    const int tid  = threadIdx.x;
    const int wave = tid >> 5;
    const int lane = tid & 31;
    if (tid < 16) rowsq[tid] = 0.0f;
    __syncthreads();

    const int mbase = blockIdx.x * 16;
    const int m  = lane & 15;               // row within tile (A), col within tile (B/C)
    const int kb = (lane >> 4) * 2;         // K sub-offset per ISA f32 A/B layout
    int mrow  = mbase + m;
    int mload = mrow < n ? mrow : (n - 1);  // clamp; clamped rows never stored

    const int n0 = (wave * 4 + 0) * 16 + m; // all < 320, always valid in Wp
    const int n1 = (wave * 4 + 1) * 16 + m;
    const int n2 = (wave * 4 + 2) * 16 + m;
    const int n3 = (wave * 4 + 3) * 16 + m;

    const float* aP  = h  + (long)mload * FDIM + kb;
    const float* b0P = Wp + (long)n0    * FDIM + kb;
    const float* b1P = Wp + (long)n1    * FDIM + kb;
    const float* b2P = Wp + (long)n2    * FDIM + kb;
    const float* b3P = Wp + (long)n3    * FDIM + kb;

    v8f acc0 = {}, acc1 = {}, acc2 = {}, acc3 = {};

    #pragma unroll 5
    for (int k0 = 0; k0 < FDIM; k0 += 4) {
        v2f a  = *(const v2f*)(aP  + k0);
        v2f b0 = *(const v2f*)(b0P + k0);
        v2f b1 = *(const v2f*)(b1P + k0);
        v2f b2 = *(const v2f*)(b2P + k0);
        v2f b3 = *(const v2f*)(b3P + k0);
        acc0 = __builtin_amdgcn_wmma_f32_16x16x4_f32(false, a, false, b0, (short)0, acc0, false, false);
        acc1 = __builtin_amdgcn_wmma_f32_16x16x4_f32(false, a, false, b1, (short)0, acc1, false, false);
        acc2 = __builtin_amdgcn_wmma_f32_16x16x4_f32(false, a, false, b2, (short)0, acc2, false, false);
        acc3 = __builtin_amdgcn_wmma_f32_16x16x4_f32(false, a, false, b3, (short)0, acc3, false, false);
    }

    // ---- epilogue (off the critical path) ----
    v8f acc[4] = {acc0, acc1, acc2, acc3};
    int ncol[4] = {n0, n1, n2, n3};

    // bias + relu (padded cols get 0 + 0 -> 0, add nothing to the norm)
    #pragma unroll
    for (int t = 0; t < 4; ++t) {
        float bn = bp[ncol[t]];
        #pragma unroll
        for (int v = 0; v < 8; ++v) {
            float cv = acc[t][v] + bn;
            acc[t][v] = cv > 0.0f ? cv : 0.0f;
        }
    }

    // squared-sum reduction: lane's VGPR v holds row (v + 8*(lane>=16)), col ncol[t]
    const int rbase = (lane >> 4) * 8;
    #pragma unroll
    for (int v = 0; v < 8; ++v) {
        float s = 0.0f;
        #pragma unroll
        for (int t = 0; t < 4; ++t) { float cv = acc[t][v]; s += cv * cv; }
        atomicAdd(&rowsq[rbase + v], s);    // ds_add_f32
    }
    __syncthreads();

    #pragma unroll
    for (int v = 0; v < 8; ++v) {
        int r = mbase + rbase + v;
        if (r < n) {
            float scale = 1.0f / fmaxf(sqrtf(rowsq[rbase + v]), 1e-12f);
            #pragma unroll
            for (int t = 0; t < 4; ++t) {
                if (ncol[t] < FDIM) out[(long)r * FDIM + ncol[t]] = acc[t][v] * scale;
            }
        }
    }
}

// ---------------- launch ----------------

extern "C" void kernel_launch(void* const* d_in, const int* in_sizes, int n_in,
                              void* d_out, int out_size, void* d_ws, size_t ws_size,
                              hipStream_t stream) {
    const float* x  = (const float*)d_in[0];
    const int*   ei = (const int*)d_in[1];
    const float* Wm = (const float*)d_in[2];
    const float* b  = (const float*)d_in[3];
    float* out = (float*)d_out;

    const int N = in_sizes[0] / FDIM;
    const int E = in_sizes[1] / 2;
    const int* row = ei;        // edge_index[0]
    const int* col = ei + E;    // edge_index[1]

    // workspace layout (bytes):
    //   dinv : [0,            N*4)
    //   deg  : [N*4,          2N*4)
    //   Wp   : [2N*4,         2N*4 + NPAD*FDIM*4)
    //   bp   : [.. ,          .. + NPAD*4)
    //   h    : [.. ,          .. + N*FDIM*4)      (16B aligned)
    char* ws = (char*)d_ws;
    size_t off = 0;
    float* dinv = (float*)(ws + off); off += (size_t)N * 4;
    int*   deg  = (int*)  (ws + off); off += (size_t)N * 4;
    float* Wp   = (float*)(ws + off); off += (size_t)NPAD * FDIM * 4;
    float* bp   = (float*)(ws + off); off += (size_t)NPAD * 4;
    off = (off + 15) & ~(size_t)15;
    float* h    = (float*)(ws + off);

    int thr = 256;
    sgc_init_deg <<<(N + thr - 1) / thr, thr, 0, stream>>>(deg, N);
    sgc_count_deg<<<(E + thr - 1) / thr, thr, 0, stream>>>(col, E, deg);
    sgc_dinv     <<<(N + thr - 1) / thr, thr, 0, stream>>>(deg, dinv, N);
    sgc_pad      <<<(NPAD * FDIM + thr - 1) / thr, thr, 0, stream>>>(Wm, b, Wp, bp);

    int hElems = N * FDIM4;
    sgc_hinit    <<<(hElems + thr - 1) / thr, thr, 0, stream>>>(x, dinv, h, N);

    // one wave32 per edge -> 8 edges per 256-thread block
    sgc_scatter  <<<(E + 7) / 8, thr, 0, stream>>>(x, row, col, dinv, h, E);

    sgc_gemm_relu_norm<<<(N + 15) / 16, 160, 0, stream>>>(h, Wp, bp, out, N);
}